// DKGModule_36971078484040
// MI455X (gfx1250) — compile-verified
//
#include <hip/hip_runtime.h>

typedef float v2f __attribute__((ext_vector_type(2)));
typedef float v8f __attribute__((ext_vector_type(8)));

#define NB 32
#define CH 256
#define GH 128
#define HH 64
#define WW 64
#define HW 4096
#define PL 256

// D = A(16x4) * B(4x16) + C, fp32 WMMA (CDNA5 V_WMMA_F32_16X16X4_F32)
__device__ __forceinline__ v8f wmma4(v2f a, v2f b, v8f c) {
  return __builtin_amdgcn_wmma_f32_16x16x4_f32(false, a, false, b, (short)0, c, false, false);
}

// ---- Kernel 1: per-(n,g) spatial mean of first-half channels; one wave each.
__global__ __launch_bounds__(256) void mean_kernel(const float* __restrict__ x,
                                                   float* __restrict__ gm) {
  int wid  = (blockIdx.x * blockDim.x + threadIdx.x) >> 5; // 0..4095 = n*128+g
  int lane = threadIdx.x & 31;
  int n = wid >> 7, g = wid & 127;
  const float4* p4 = (const float4*)(x + ((size_t)n * CH + g) * HW);
  float s = 0.f;
  for (int i = lane; i < HW / 4; i += 32) {
    float4 v = p4[i];
    s += v.x + v.y + v.z + v.w;
  }
  for (int off = 16; off > 0; off >>= 1) s += __shfl_down(s, off, 32);
  if (lane == 0) gm[wid] = s * (1.f / (float)HW);
}

// ---- Kernel 2: transpose w_static (go,gi,ky,kx) -> wt[tap][go][gi] for coalesced staging.
__global__ void wprep_kernel(const float* __restrict__ wsrc, float* __restrict__ wt) {
  int i = blockIdx.x * 256 + threadIdx.x;
  if (i < GH * GH * 9) {
    int tap = i / (GH * GH);
    int r   = i - tap * (GH * GH); // r = go*128 + gi
    wt[i] = wsrc[r * 9 + tap];
  }
}

// ---- Kernel 3: one block per (n, output row h). 8 waves.
//   Phase A: y_res tile -> sY, fuse GEMM K=0..127
//   Phase B: static conv (WMMA, 9 taps) -> sY, fuse GEMM K=128..255
__global__ __launch_bounds__(256) void fused_kernel(
    const float* __restrict__ x,
    const float* __restrict__ w_kernel, const float* __restrict__ b_kernel,
    const float* __restrict__ w_k, const float* __restrict__ b_k,
    const float* __restrict__ b_static,
    const float* __restrict__ w_fuse, const float* __restrict__ b_fuse,
    const float* __restrict__ gm, const float* __restrict__ wt,
    float* __restrict__ out)
{
  __shared__ float sX[66 * 36];   // [px+1][gi]  (gi stride 36 -> conflict-free v2f)
  __shared__ float sW[128 * 36];  // [go][gi]
  __shared__ float sY[64 * 132];  // [px][c]     (c stride 132)

  const int n    = blockIdx.x >> 6;
  const int h    = blockIdx.x & 63;
  const int t    = threadIdx.x;
  const int lane = t & 31;
  const int wv   = t >> 5;       // 0..7
  const int l15  = lane & 15;    // column / M-row selector within tile
  const int dl   = lane >> 4;    // K-half selector (A/B layout), row+8 selector (C/D)
  const int oB   = (wv >> 1) * 64; // fuse-GEMM output-channel base (4 M-tiles)
  const int nB   = (wv & 1) * 32;  // pixel base (2 N-tiles)
  const int goB  = (wv >> 1) * 32; // conv go base (2 M-tiles)

  const v8f vzero = {0.f,0.f,0.f,0.f,0.f,0.f,0.f,0.f};
  v8f accO[4][2]; // fuse output 256x64 across 8 waves
  v8f accS[2][2]; // static conv 128x64 across 8 waves
  #pragma unroll
  for (int a = 0; a < 4; ++a)
    #pragma unroll
    for (int b = 0; b < 2; ++b) accO[a][b] = vzero;
  #pragma unroll
  for (int a = 0; a < 2; ++a)
    #pragma unroll
    for (int b = 0; b < 2; ++b) accS[a][b] = vzero;

  // ================= Phase A: dynamic depthwise conv -> sY[px][c] =================
  float ka[9], kb[9];
  {
    float wkv = w_k[0], bkv = b_k[0];
    #pragma unroll
    for (int tp = 0; tp < 9; ++tp) {
      ka[tp] = wkv * w_kernel[tp];
      kb[tp] = wkv * b_kernel[tp] + bkv;
    }
  }
  for (int i = 0; i < 32; ++i) {
    int e  = i * 256 + t;     // 0..8191 over (c,px)
    int c  = e >> 6;          // 0..127
    int px = e & 63;
    float g = gm[n * GH + c];
    const float* xb = x + ((size_t)n * CH + c) * HW;
    float acc = 0.f;
    #pragma unroll
    for (int dy = 0; dy < 3; ++dy) {
      int r = h + dy - 1;
      if (r < 0 || r >= HH) continue;
      #pragma unroll
      for (int dx = 0; dx < 3; ++dx) {
        int q = px + dx - 1;
        if (q < 0 || q >= WW) continue;
        int tp = dy * 3 + dx;
        acc += (g * ka[tp] + kb[tp]) * xb[r * WW + q];
      }
    }
    sY[px * 132 + c] = acc;
  }
  __syncthreads();

  // ---- Fuse GEMM chunk 1: out += w_fuse[:, 0:128] @ y_res
  for (int k0 = 0; k0 < 128; k0 += 4) {
    int kk = k0 + 2 * dl;
    v2f aF[4], bF[2];
    #pragma unroll
    for (int mt = 0; mt < 4; ++mt)
      aF[mt] = *(const v2f*)(w_fuse + (size_t)(oB + mt * 16 + l15) * CH + kk);
    #pragma unroll
    for (int nt = 0; nt < 2; ++nt)
      bF[nt] = *(const v2f*)(&sY[(nB + nt * 16 + l15) * 132 + kk]);
    #pragma unroll
    for (int mt = 0; mt < 4; ++mt)
      #pragma unroll
      for (int nt = 0; nt < 2; ++nt)
        accO[mt][nt] = wmma4(aF[mt], bF[nt], accO[mt][nt]);
  }

  // ================= Phase B: static 3x3 conv via WMMA (9 taps) =================
  for (int dy = 0; dy < 3; ++dy) {
    int r = h + dy - 1;
    for (int gi0 = 0; gi0 < 128; gi0 += 32) {
      // stage zero-padded input row chunk: sX[px+1][gi], px = -1..64, gi = gi0..gi0+31
      for (int i = t; i < 66 * 32; i += 256) {
        int gi  = i / 66;
        int pxp = i - gi * 66;
        int q   = pxp - 1;
        float v = 0.f;
        if (r >= 0 && r < HH && q >= 0 && q < WW)
          v = x[((size_t)n * CH + GH + gi0 + gi) * HW + r * WW + q];
        sX[pxp * 36 + gi] = v;
      }
      for (int dx = 0; dx < 3; ++dx) {
        // stage weight chunk for this tap: sW[go][gi-gi0]
        const float* wtp = wt + (size_t)(dy * 3 + dx) * (GH * GH) + gi0;
        for (int i = t; i < 128 * 32; i += 256) {
          int go  = i >> 5;
          int gic = i & 31;
          sW[go * 36 + gic] = wtp[go * GH + gic];
        }
        __syncthreads(); // sX + sW visible to all
        for (int k0 = 0; k0 < 32; k0 += 4) {
          int kk = k0 + 2 * dl;
          v2f aF[2], bF[2];
          #pragma unroll
          for (int gt = 0; gt < 2; ++gt)
            aF[gt] = *(const v2f*)(&sW[(goB + gt * 16 + l15) * 36 + kk]);
          #pragma unroll
          for (int nt = 0; nt < 2; ++nt)
            bF[nt] = *(const v2f*)(&sX[(nB + nt * 16 + l15 + dx) * 36 + kk]);
          #pragma unroll
          for (int gt = 0; gt < 2; ++gt)
            #pragma unroll
            for (int nt = 0; nt < 2; ++nt)
              accS[gt][nt] = wmma4(aF[gt], bF[nt], accS[gt][nt]);
        }
        __syncthreads(); // done reading before next sW/sX overwrite
      }
    }
  }

  // write y_stat (+ b_static) into sY[px][go] using documented C-layout
  #pragma unroll
  for (int gt = 0; gt < 2; ++gt)
    #pragma unroll
    for (int nt = 0; nt < 2; ++nt)
      #pragma unroll
      for (int v = 0; v < 8; ++v) {
        int go = goB + gt * 16 + v + 8 * dl;
        int px = nB + nt * 16 + l15;
        sY[px * 132 + go] = accS[gt][nt][v] + b_static[go];
      }
  __syncthreads();

  // ---- Fuse GEMM chunk 2: out += w_fuse[:, 128:256] @ y_stat
  for (int k0 = 0; k0 < 128; k0 += 4) {
    int kk = k0 + 2 * dl;
    v2f aF[4], bF[2];
    #pragma unroll
    for (int mt = 0; mt < 4; ++mt)
      aF[mt] = *(const v2f*)(w_fuse + (size_t)(oB + mt * 16 + l15) * CH + 128 + kk);
    #pragma unroll
    for (int nt = 0; nt < 2; ++nt)
      bF[nt] = *(const v2f*)(&sY[(nB + nt * 16 + l15) * 132 + kk]);
    #pragma unroll
    for (int mt = 0; mt < 4; ++mt)
      #pragma unroll
      for (int nt = 0; nt < 2; ++nt)
        accO[mt][nt] = wmma4(aF[mt], bF[nt], accO[mt][nt]);
  }

  // ---- store out[n][o][h][px] = accO + b_fuse
  #pragma unroll
  for (int mt = 0; mt < 4; ++mt)
    #pragma unroll
    for (int nt = 0; nt < 2; ++nt)
      #pragma unroll
      for (int v = 0; v < 8; ++v) {
        int o  = oB + mt * 16 + v + 8 * dl;
        int px = nB + nt * 16 + l15;
        out[((size_t)n * PL + o) * HW + h * WW + px] = accO[mt][nt][v] + b_fuse[o];
      }
}

extern "C" void kernel_launch(void* const* d_in, const int* in_sizes, int n_in,
                              void* d_out, int out_size, void* d_ws, size_t ws_size,
                              hipStream_t stream) {
  const float* x        = (const float*)d_in[0];
  const float* w_kernel = (const float*)d_in[1];
  const float* b_kernel = (const float*)d_in[2];
  const float* w_k      = (const float*)d_in[3];
  const float* b_k      = (const float*)d_in[4];
  const float* w_static = (const float*)d_in[5];
  const float* b_static = (const float*)d_in[6];
  const float* w_fuse   = (const float*)d_in[7];
  const float* b_fuse   = (const float*)d_in[8];
  float* out = (float*)d_out;

  float* gm  = (float*)d_ws;          // 4096 floats: per-(n,g) means
  float* wtr = gm + NB * GH;          // 147456 floats: transposed w_static

  // 4096 waves (one per (n,g)) = 512 blocks * 8 waves
  mean_kernel<<<dim3(512), dim3(256), 0, stream>>>(x, gm);
  wprep_kernel<<<dim3((GH * GH * 9 + 255) / 256), dim3(256), 0, stream>>>(w_static, wtr);
  fused_kernel<<<dim3(NB * HH), dim3(256), 0, stream>>>(
      x, w_kernel, b_kernel, w_k, b_k, b_static, w_fuse, b_fuse, gm, wtr, out);
}